// DirectionalLoss_30451318129051
// MI455X (gfx1250) — compile-verified
//
#include <hip/hip_runtime.h>
#include <math.h>

typedef __attribute__((ext_vector_type(2))) float v2f;
typedef __attribute__((ext_vector_type(8))) float v8f;

#define H_DIM 4096
#define TPB 256
#define NSLOT 64          // accumulator slot sets to de-serialize f64 atomics
#define SLOT_STRIDE 4     // 3 doubles used, padded to 32B
#define EPS 1e-6f

// ---------------------------------------------------------------- zero init
__global__ void DL_zero(double* __restrict__ acc) {
    int i = blockIdx.x * blockDim.x + threadIdx.x;
    if (i < NSLOT * SLOT_STRIDE) acc[i] = 0.0;
}

// ---------------------------------------------------------------- main pass
__global__ __launch_bounds__(TPB) void DL_rows(const float* __restrict__ pred,
                                               const float* __restrict__ targ,
                                               double* __restrict__ acc) {
    const int row  = blockIdx.x;
    const int tid  = threadIdx.x;
    const size_t rowOff = (size_t)row * H_DIM;
    const float* __restrict__ prow = pred + rowOff;
    const float* __restrict__ trow = targ + rowOff;
    const int base = tid * 16;

    // ---- stream 16 contiguous elements per thread via b128 loads ----
    float pv[17], tv[17];
    const float4* p4 = (const float4*)(prow + base);
    const float4* t4 = (const float4*)(trow + base);
#pragma unroll
    for (int i = 0; i < 4; ++i) {
        float4 a = p4[i];
        pv[4*i+0] = a.x; pv[4*i+1] = a.y; pv[4*i+2] = a.z; pv[4*i+3] = a.w;
        float4 b = t4[i];
        tv[4*i+0] = b.x; tv[4*i+1] = b.y; tv[4*i+2] = b.z; tv[4*i+3] = b.w;
    }
    const int nidx = (base + 16 < H_DIM) ? (base + 16) : (H_DIM - 1);
    pv[16] = prow[nidx];
    tv[16] = trow[nidx];

    // ---- per-thread partials (all fused in one pass) ----
    float sp = 0.f, st = 0.f, spp = 0.f, stt = 0.f, spt = 0.f, ssq = 0.f, mcnt = 0.f;
#pragma unroll
    for (int k = 0; k < 16; ++k) {
        float p = pv[k], t = tv[k];
        sp += p; st += t;
        spp = fmaf(p, p, spp);
        stt = fmaf(t, t, stt);
        spt = fmaf(p, t, spt);
        float d = p - t;
        ssq = fmaf(d, d, ssq);
    }
    const int kmax = (base + 16 <= H_DIM - 1) ? 16 : (H_DIM - 1 - base);
#pragma unroll
    for (int k = 0; k < 16; ++k) {
        if (k < kmax) {
            float dp = pv[k + 1] - pv[k];
            float dt = tv[k + 1] - tv[k];
            bool m = (dt == 0.f) ||
                     (((dp > 0.f) == (dt > 0.f)) && ((dp < 0.f) == (dt < 0.f)));
            mcnt += m ? 1.f : 0.f;
        }
    }

    // ---- wave32 shuffle-tree reduction (gfx1250 is wave32-only) ----
#pragma unroll
    for (int off = 16; off > 0; off >>= 1) {
        sp   += __shfl_xor(sp,   off, 32);
        st   += __shfl_xor(st,   off, 32);
        spp  += __shfl_xor(spp,  off, 32);
        stt  += __shfl_xor(stt,  off, 32);
        spt  += __shfl_xor(spt,  off, 32);
        ssq  += __shfl_xor(ssq,  off, 32);
        mcnt += __shfl_xor(mcnt, off, 32);
    }

    // ---- cross-wave reduction via V_WMMA_F32_16X16X4_F32 ----
    // LDS layout: lds[k*16 + m] = partial of quantity m from wave k (A-matrix rows = quantities)
    __shared__ float lds[8 * 16];
    if (tid < 8 * 16) lds[tid] = 0.f;
    __syncthreads();
    const int wid  = tid >> 5;
    const int lane = tid & 31;
    if (lane == 0) {
        float q[7] = { sp, st, spp, stt, spt, ssq, mcnt };
#pragma unroll
        for (int i = 0; i < 7; ++i) lds[wid * 16 + i] = q[i];
    }
    __syncthreads();

    if (wid == 0) {  // uniform branch: EXEC all-ones within wave 0 (WMMA requirement)
        const int m  = lane & 15;
        const int hi = lane >> 4;
        // 32-bit A 16x4 layout: VGPR0 = K=0 (lanes 0-15) / K=2 (lanes 16-31); VGPR1 = K=1 / K=3
        v2f a0, a1, bmat;
        a0.x = lds[(2 * hi + 0) * 16 + m];
        a0.y = lds[(2 * hi + 1) * 16 + m];
        a1.x = lds[(2 * hi + 4) * 16 + m];
        a1.y = lds[(2 * hi + 5) * 16 + m];
        bmat.x = 1.0f; bmat.y = 1.0f;          // B = ones(4x16) -> D[m,n] = sum_k A[m,k]
        v8f c = {0.f, 0.f, 0.f, 0.f, 0.f, 0.f, 0.f, 0.f};
        c = __builtin_amdgcn_wmma_f32_16x16x4_f32(false, a0, false, bmat, (short)0, c, false, false);
        c = __builtin_amdgcn_wmma_f32_16x16x4_f32(false, a1, false, bmat, (short)0, c, false, false);

        if (lane == 0) {  // D[m,0] lives in lane 0, VGPR m (m = 0..7)
            float S_p  = c[0], S_t  = c[1], S_pp = c[2], S_tt = c[3];
            float S_pt = c[4], S_sq = c[5], S_m  = c[6];
            const float Hf = (float)H_DIM;
            float mp   = S_p / Hf, mt = S_t / Hf;
            float varp = (S_pp - S_p * mp) / (Hf - 1.f);   // unbiased (ddof=1)
            float vart = (S_tt - S_t * mt) / (Hf - 1.f);
            float sdp  = sqrtf(fmaxf(varp, 0.f));
            float sdt  = sqrtf(fmaxf(vart, 0.f));
            float cov  = S_pt / Hf - mp * mt;
            float r    = cov / ((sdp + EPS) * (sdt + EPS));

            double* slot = acc + (size_t)(blockIdx.x & (NSLOT - 1)) * SLOT_STRIDE;
            atomicAdd(&slot[0], (double)S_sq);  // global MSE numerator
            atomicAdd(&slot[1], (double)S_m);   // global sign-match count
            atomicAdd(&slot[2], (double)r);     // sum of per-row correlations
        }
    }
}

// ---------------------------------------------------------------- finalize
__global__ void DL_finalize(const double* __restrict__ acc, float* __restrict__ out, int B) {
    double ssq = 0.0, match = 0.0, corr = 0.0;
    for (int s = 0; s < NSLOT; ++s) {
        ssq   += acc[s * SLOT_STRIDE + 0];
        match += acc[s * SLOT_STRIDE + 1];
        corr  += acc[s * SLOT_STRIDE + 2];
    }
    double nElem   = (double)B * (double)H_DIM;
    double mse     = ssq / nElem;
    double dirAcc  = match / ((double)B * (double)(H_DIM - 1));
    double rMean   = corr / (double)B;
    double dirLoss = 1.0 - dirAcc;
    double corLoss = 0.5 * (1.0 - rMean);
    out[0] = (float)(0.5 * mse + 0.5 * 0.5 * (dirLoss + corLoss));
}

// ---------------------------------------------------------------- launcher
extern "C" void kernel_launch(void* const* d_in, const int* in_sizes, int n_in,
                              void* d_out, int out_size, void* d_ws, size_t ws_size,
                              hipStream_t stream) {
    const float* pred = (const float*)d_in[0];
    const float* targ = (const float*)d_in[1];
    float* out  = (float*)d_out;
    double* acc = (double*)d_ws;
    const int B = in_sizes[0] / H_DIM;   // 8192

    DL_zero<<<1, NSLOT * SLOT_STRIDE, 0, stream>>>(acc);
    DL_rows<<<B, TPB, 0, stream>>>(pred, targ, acc);
    DL_finalize<<<1, 1, 0, stream>>>(acc, out, B);
}